// ClassWiseRegressionLoss_14181982011577
// MI455X (gfx1250) — compile-verified
//
#include <hip/hip_runtime.h>

typedef float v2f __attribute__((ext_vector_type(2)));
typedef float v8f __attribute__((ext_vector_type(8)));

#define TPB 256

__device__ __forceinline__ float smooth_l1(float d) {
    float ad = __builtin_fabsf(d);
    return (ad < 1.0f) ? (0.5f * d * d) : (ad - 0.5f);
}

// Wave+block reduction. Each lane contributes two f32 accumulators.
// A = ones (16x4 f32), B = the wave's 64 accumulator registers (4x16 f32):
// D[m,n] = sum_k B[k,n]  -- an exact f32 4:1 add tree in one v_wmma.
// d[0] on lane L holds column-sum(L%16); a 16-lane butterfly then sums the 16
// columns (covering every accumulator exactly once, independent of the exact
// B-operand lane/VGPR bijection). LDS folds the 8 waves of the block.
__device__ __forceinline__ float block_reduce(float acc0, float acc1, float* lds) {
    v2f a; a[0] = 1.0f; a[1] = 1.0f;
    v2f b; b[0] = acc0; b[1] = acc1;
    v8f c = {};
    v8f d = __builtin_amdgcn_wmma_f32_16x16x4_f32(
        /*neg_a=*/false, a, /*neg_b=*/false, b,
        /*c_mod=*/(short)0, c, /*reuse_a=*/false, /*reuse_b=*/false);
    float t = d[0];
    t += __shfl_xor(t, 1, 32);
    t += __shfl_xor(t, 2, 32);
    t += __shfl_xor(t, 4, 32);
    t += __shfl_xor(t, 8, 32);
    const int wave = threadIdx.x >> 5;
    if ((threadIdx.x & 31) == 0) lds[wave] = t;
    __syncthreads();
    float r = 0.0f;
    if (threadIdx.x == 0) {
        const int nw = (int)(blockDim.x >> 5);
        for (int w = 0; w < nw; ++w) r += lds[w];
    }
    return r;
}

__global__ __launch_bounds__(TPB) void cwr_partial_kernel(
    const float* __restrict__ pred, const int* __restrict__ labels,
    const float* __restrict__ targets, float* __restrict__ partials,
    int n, int c2) {
    __shared__ float lds[TPB / 32];
    const int stride = (int)(gridDim.x * TPB);
    const int tid = (int)(blockIdx.x * TPB + threadIdx.x);
    float acc0 = 0.0f, acc1 = 0.0f;
    for (int i = tid; i < n; i += 2 * stride) {
        {
            const int lab = labels[i] - 1;  // 1-based class id -> 0-based
            const float2 cp = *(const float2*)(pred + (size_t)i * (size_t)c2 + (size_t)(lab << 1));
            const float2 tg = *(const float2*)(targets + ((size_t)i << 1));
            acc0 += smooth_l1(cp.x - tg.x) + smooth_l1(cp.y - tg.y);
        }
        const int j = i + stride;
        if (j < n) {
            const int lab = labels[j] - 1;
            const float2 cp = *(const float2*)(pred + (size_t)j * (size_t)c2 + (size_t)(lab << 1));
            const float2 tg = *(const float2*)(targets + ((size_t)j << 1));
            acc1 += smooth_l1(cp.x - tg.x) + smooth_l1(cp.y - tg.y);
        }
    }
    const float r = block_reduce(acc0, acc1, lds);
    if (threadIdx.x == 0) partials[blockIdx.x] = r;
}

__global__ __launch_bounds__(TPB) void cwr_final_kernel(
    const float* __restrict__ partials, float* __restrict__ out,
    int n_partials, float inv_scale) {
    __shared__ float lds[TPB / 32];
    const int tid = (int)threadIdx.x;
    float acc0 = 0.0f, acc1 = 0.0f;
    for (int i = tid; i < n_partials; i += 2 * TPB) {
        acc0 += partials[i];
        if (i + TPB < n_partials) acc1 += partials[i + TPB];
    }
    const float r = block_reduce(acc0, acc1, lds);
    if (tid == 0) out[0] = r * inv_scale;
}

extern "C" void kernel_launch(void* const* d_in, const int* in_sizes, int n_in,
                              void* d_out, int out_size, void* d_ws, size_t ws_size,
                              hipStream_t stream) {
    const float* pred    = (const float*)d_in[0];  // [N, C, 2] f32
    const int*   labels  = (const int*)d_in[1];    // [N] (1-based class ids)
    const float* targets = (const float*)d_in[2];  // [N, 2] f32
    float* out = (float*)d_out;
    float* partials = (float*)d_ws;

    const int n  = in_sizes[1];            // N = 262144
    const int c2 = in_sizes[0] / n;        // C*2 = 400 floats per row

    // Two rows per thread -> N/512 blocks (= 512 for N=262144); cap so the
    // final single-block kernel's partial array stays small (grid-stride
    // covers any remainder).
    int blocks = (n + 2 * TPB - 1) / (2 * TPB);
    if (blocks > 2 * TPB) blocks = 2 * TPB;
    if (blocks < 1) blocks = 1;

    cwr_partial_kernel<<<blocks, TPB, 0, stream>>>(pred, labels, targets, partials, n, c2);
    // mean over 2N elements, scaled by 2  =>  total / N
    cwr_final_kernel<<<1, TPB, 0, stream>>>(partials, out, blocks, 1.0f / (float)n);
}